// GNN_73512660238465
// MI455X (gfx1250) — compile-verified
//
#include <hip/hip_runtime.h>
#include <hip/hip_bf16.h>
#include <math.h>

#define NNODES 50000
#define NB     100
#define NPG    500
#define NEDGE  600000
#define NC128  128
#define TOPK   32

typedef __attribute__((ext_vector_type(2))) float v2f;
typedef __attribute__((ext_vector_type(8))) float v8f;

__device__ __forceinline__ float lrelu(float x) { return x > 0.f ? x : 0.01f * x; }

// ---------------------------------------------------------------------------
// WMMA fp32 GEMM: out[M,128] = A[M,128] @ W[128,128]
// One wave -> 16 rows x 128 cols; W staged in 64KB LDS via async-to-LDS DMA.
// ---------------------------------------------------------------------------
__global__ __launch_bounds__(128) void k_gemm128(const float* __restrict__ A,
                                                 const float* __restrict__ W,
                                                 float* __restrict__ out, int M) {
    __shared__ float lw[128 * 128];

#if defined(__gfx1250__) && __has_builtin(__builtin_amdgcn_global_load_async_to_lds_b128)
    {
        // 64KB straight copy: per-lane 16B async transfers, tracked on ASYNCcnt
        typedef int v4i_ __attribute__((vector_size(16)));
        typedef __attribute__((address_space(1))) v4i_* g4p;
        typedef __attribute__((address_space(3))) v4i_* l4p;
        const __attribute__((address_space(1))) char* gp =
            (const __attribute__((address_space(1))) char*)W;
        __attribute__((address_space(3))) char* lp =
            (__attribute__((address_space(3))) char*)lw;
        for (int i = threadIdx.x; i < 4096; i += 128)
            __builtin_amdgcn_global_load_async_to_lds_b128(
                (g4p)(gp + (size_t)i * 16), (l4p)(lp + i * 16), 0, 0);
#if __has_builtin(__builtin_amdgcn_s_wait_asynccnt)
        __builtin_amdgcn_s_wait_asynccnt(0);
#else
        asm volatile("s_wait_asynccnt 0x0" ::: "memory");
#endif
    }
#else
    for (int i = threadIdx.x; i < 4096; i += 128)
        ((float4*)lw)[i] = ((const float4*)W)[i];
#endif
    __syncthreads();

    int wave = threadIdx.x >> 5;
    int lane = threadIdx.x & 31;
    int tile = blockIdx.x * 4 + wave;
    if (tile * 16 >= M) return;                 // wave-uniform -> EXEC stays all-1s

    int row0 = tile * 16;
    int lr   = lane & 15;
    int kh   = (lane >> 4) << 1;                // 0 for lanes 0-15, 2 for lanes 16-31
    const float* arow = A + (size_t)(row0 + lr) * 128;

    v8f acc[8] = {};
    for (int kk = 0; kk < 32; ++kk) {
        int k0 = kk * 4;
        // A fragment, 16x4 f32 layout: VGPR0=K(kh), VGPR1=K(kh+1)
        v2f a = *(const v2f*)(arow + k0 + kh);
#pragma unroll
        for (int nt = 0; nt < 8; ++nt) {
            int col = nt * 16 + lr;
            v2f b;
            b.x = lw[(k0 + kh) * 128 + col];
            b.y = lw[(k0 + kh + 1) * 128 + col];
            acc[nt] = __builtin_amdgcn_wmma_f32_16x16x4_f32(
                false, a, false, b, (short)0, acc[nt], false, false);
        }
    }
    // D layout: VGPR v -> M = v + (lane<16 ? 0 : 8), N = lane&15
    int mbase = (lane >> 4) * 8;
#pragma unroll
    for (int nt = 0; nt < 8; ++nt) {
        int col = nt * 16 + lr;
#pragma unroll
        for (int v = 0; v < 8; ++v)
            out[(size_t)(row0 + mbase + v) * 128 + col] = acc[nt][v];
    }
}

// ---------------------------------------------------------------------------
// Degree / normalization
// ---------------------------------------------------------------------------
__global__ void k_zero(float* __restrict__ p, int n) {
    int i = blockIdx.x * blockDim.x + threadIdx.x;
    if (i < n) p[i] = 0.f;
}
__global__ void k_deg(const int* __restrict__ dst, float* __restrict__ deg, int e) {
    int i = blockIdx.x * blockDim.x + threadIdx.x;
    if (i < e) unsafeAtomicAdd(&deg[dst[i]], 1.0f);
}
__global__ void k_rsqrt(float* __restrict__ d, int n) {
    int i = blockIdx.x * blockDim.x + threadIdx.x;
    if (i < n) d[i] = rsqrtf(1.0f + d[i]);
}

// ---------------------------------------------------------------------------
// GCN aggregate: out = h*dis^2 + b, then scatter msgs, then tanh
// ---------------------------------------------------------------------------
__global__ void k_gcn_init(const float* __restrict__ h, const float* __restrict__ dis,
                           const float* __restrict__ b, float* __restrict__ out, int n) {
    int i = blockIdx.x * blockDim.x + threadIdx.x;
    if (i >= n) return;
    int node = i >> 7, c = i & 127;
    float d = dis[node];
    out[i] = h[i] * d * d + b[c];
}
__global__ void k_scatter(const float* __restrict__ h, const int* __restrict__ src,
                          const int* __restrict__ dst, const float* __restrict__ dis,
                          float* __restrict__ out, int e) {
    int gid = blockIdx.x * blockDim.x + threadIdx.x;
    int ed = gid >> 5;
    if (ed >= e) return;
    int lane = gid & 31;
    int s = src[ed], d = dst[ed];
    float w = dis[s] * dis[d];
    float4 v = *(const float4*)(h + (size_t)s * 128 + lane * 4);
    float* o = out + (size_t)d * 128 + lane * 4;
    unsafeAtomicAdd(o + 0, v.x * w);
    unsafeAtomicAdd(o + 1, v.y * w);
    unsafeAtomicAdd(o + 2, v.z * w);
    unsafeAtomicAdd(o + 3, v.w * w);
}
__global__ void k_tanh(float* __restrict__ p, int n) {
    int i = blockIdx.x * blockDim.x + threadIdx.x;
    if (i < n) p[i] = tanhf(p[i]);
}

// ---------------------------------------------------------------------------
// Color GCN (128 -> 1)
// ---------------------------------------------------------------------------
__global__ void k_colorH(const float* __restrict__ g, const float* __restrict__ W,
                         float* __restrict__ out, int n) {
    int i = blockIdx.x * blockDim.x + threadIdx.x;
    if (i >= n) return;
    const float* row = g + (size_t)i * 128;
    float acc = 0.f;
#pragma unroll 4
    for (int c = 0; c < 128; ++c) acc += row[c] * W[c];
    out[i] = acc;
}
__global__ void k_color_init(const float* __restrict__ hc, const float* __restrict__ dis,
                             const float* __restrict__ b, float* __restrict__ out, int n) {
    int i = blockIdx.x * blockDim.x + threadIdx.x;
    if (i >= n) return;
    float d = dis[i];
    out[i] = hc[i] * d * d + b[0];
}
__global__ void k_color_scatter(const float* __restrict__ hc, const int* __restrict__ src,
                                const int* __restrict__ dst, const float* __restrict__ dis,
                                float* __restrict__ out, int e) {
    int i = blockIdx.x * blockDim.x + threadIdx.x;
    if (i >= e) return;
    int s = src[i], d = dst[i];
    unsafeAtomicAdd(&out[d], hc[s] * dis[s] * dis[d]);
}

// ---------------------------------------------------------------------------
// Per-graph stable top-32 by score (descending; ties -> lower index first,
// matching stable argsort of -score). One block per graph.
// ---------------------------------------------------------------------------
__global__ __launch_bounds__(512) void k_topk(const float* __restrict__ score,
                                              int* __restrict__ topIdx) {
    __shared__ float sv[512];
    __shared__ float rv[512];
    __shared__ int   ri[512];
    int b = blockIdx.x, t = threadIdx.x;
    sv[t] = (t < NPG) ? score[b * NPG + t] : -3.4e38f;
    __syncthreads();
    for (int k = 0; k < TOPK; ++k) {
        rv[t] = sv[t]; ri[t] = t;
        __syncthreads();
        for (int s = 256; s > 0; s >>= 1) {
            if (t < s) {
                float v2 = rv[t + s]; int i2 = ri[t + s];
                if (v2 > rv[t] || (v2 == rv[t] && i2 < ri[t])) { rv[t] = v2; ri[t] = i2; }
            }
            __syncthreads();
        }
        if (t == 0) {
            topIdx[b * TOPK + k] = b * NPG + ri[0];
            sv[ri[0]] = -3.4e38f;
        }
        __syncthreads();
    }
}

// ---------------------------------------------------------------------------
// Sort-pool einsum: h[b,c,k] = sum_d pooled[b,k,d]*W1[c,d] + b1[c]
// d spans concat over streams j<=i of (feat[j][0..3] (128 each), color[j] (1))
// ---------------------------------------------------------------------------
struct FeatPtrs { const float* f[16]; const float* c[4]; };

__global__ __launch_bounds__(512) void k_einsum(FeatPtrs P, const int* __restrict__ topIdx,
                                                const float* __restrict__ W1,
                                                const float* __restrict__ b1,
                                                float* __restrict__ hout,
                                                int ncat, int D) {
    int b = blockIdx.x, t = threadIdx.x;
    int c = t >> 5, k = t & 31;
    int node = topIdx[b * TOPK + k];
    const float* wrow = W1 + (size_t)c * D;
    float acc = b1[c];
    for (int j = 0; j < ncat; ++j) {
        const float* wj = wrow + j * 513;
        for (int tt = 0; tt < 4; ++tt) {
            const float* fv = P.f[j * 4 + tt] + (size_t)node * 128;
            const float* wv = wj + tt * 128;
#pragma unroll 4
            for (int cc = 0; cc < 128; ++cc) acc += fv[cc] * wv[cc];
        }
        acc += P.c[j][node] * wj[512];
    }
    hout[b * 512 + c * 32 + k] = acc;
}

// lrelu -> avgpool(2)+lrelu -> conv1d(16->32, K=5, valid) + lrelu -> flatten
__global__ __launch_bounds__(512) void k_conv(const float* __restrict__ hin,
                                              const float* __restrict__ W2,
                                              const float* __restrict__ b2,
                                              float* __restrict__ z, int ipart) {
    __shared__ float l1[512];
    __shared__ float l2[256];
    int b = blockIdx.x, t = threadIdx.x;
    l1[t] = lrelu(hin[b * 512 + t]);
    __syncthreads();
    if (t < 256) {
        int c = t >> 4, p = t & 15;
        float m = 0.5f * (l1[c * 32 + 2 * p] + l1[c * 32 + 2 * p + 1]);
        l2[c * 16 + p] = lrelu(m);
    }
    __syncthreads();
    if (t < 384) {
        int o = t / 12, p = t % 12;
        float acc = b2[o];
        for (int ic = 0; ic < 16; ++ic)
#pragma unroll
            for (int kk = 0; kk < 5; ++kk)
                acc += l2[ic * 16 + p + kk] * W2[o * 80 + ic * 5 + kk];
        z[b * 1536 + ipart * 384 + t] = lrelu(acc);
    }
}

// z = lrelu(z@W+b), applied nlr times total
__global__ void k_dense(const float* __restrict__ in, const float* __restrict__ W,
                        const float* __restrict__ bias, float* __restrict__ out,
                        int M, int K, int Nc, int nlr) {
    int idx = blockIdx.x * blockDim.x + threadIdx.x;
    if (idx >= M * Nc) return;
    int m = idx / Nc, n = idx % Nc;
    float acc = bias[n];
    const float* a = in + (size_t)m * K;
    for (int k = 0; k < K; ++k) acc += a[k] * W[(size_t)k * Nc + n];
    acc = lrelu(acc);
    if (nlr > 1) acc = lrelu(acc);
    out[idx] = acc;
}

// Label-smoothing cross-entropy over [100,10] logits
__global__ __launch_bounds__(128) void k_loss(const float* __restrict__ logits,
                                              const int* __restrict__ y,
                                              float* __restrict__ lossOut) {
    __shared__ float s_nll[128];
    __shared__ float s_sum[128];
    int t = threadIdx.x;
    float nll = 0.f, tot = 0.f;
    if (t < NB) {
        const float* row = logits + t * 10;
        float mx = row[0];
        for (int c = 1; c < 10; ++c) mx = fmaxf(mx, row[c]);
        float se = 0.f;
        for (int c = 0; c < 10; ++c) se += expf(row[c] - mx);
        float lse = mx + logf(se);
        for (int c = 0; c < 10; ++c) tot += row[c] - lse;
        nll = -(row[y[t]] - lse);
    }
    s_nll[t] = nll; s_sum[t] = tot;
    __syncthreads();
    for (int s = 64; s > 0; s >>= 1) {
        if (t < s) { s_nll[t] += s_nll[t + s]; s_sum[t] += s_sum[t + s]; }
        __syncthreads();
    }
    if (t == 0) {
        float nllm   = s_nll[0] / 100.f;
        float lpmean = s_sum[0] / 1000.f;
        lossOut[0] = 0.1f * (lpmean / 10.f) + 0.9f * nllm;
    }
}

// ---------------------------------------------------------------------------
extern "C" void kernel_launch(void* const* d_in, const int* in_sizes, int n_in,
                              void* d_out, int out_size, void* d_ws, size_t ws_size,
                              hipStream_t stream) {
    (void)in_sizes; (void)n_in; (void)out_size; (void)ws_size;
    const float* x        = (const float*)d_in[0];
    const int*   edge_src = (const int*)d_in[2];
    const int*   edge_dst = (const int*)d_in[3];
    const int*   y        = (const int*)d_in[5];
    const float* gcn_W[4][3]; const float* gcn_b[4][3];
    {
        int p = 6;
        for (int blk = 0; blk < 4; ++blk) for (int i = 0; i < 3; ++i) gcn_W[blk][i] = (const float*)d_in[p++];
        for (int blk = 0; blk < 4; ++blk) for (int i = 0; i < 3; ++i) gcn_b[blk][i] = (const float*)d_in[p++];
    }
    const float* color_W[4]; const float* color_b[4];
    for (int i = 0; i < 4; ++i) { color_W[i] = (const float*)d_in[30 + i]; color_b[i] = (const float*)d_in[34 + i]; }
    const float* Wv[3]; const float* Wo[3];
    for (int i = 0; i < 3; ++i) { Wv[i] = (const float*)d_in[38 + i]; Wo[i] = (const float*)d_in[41 + i]; }
    const float* conv_W1[4]; const float* conv_b1[4]; const float* conv_W2[4]; const float* conv_b2[4];
    for (int i = 0; i < 4; ++i) {
        conv_W1[i] = (const float*)d_in[48 + i]; conv_b1[i] = (const float*)d_in[52 + i];
        conv_W2[i] = (const float*)d_in[56 + i]; conv_b2[i] = (const float*)d_in[60 + i];
    }
    const float* cls_W[3]; const float* cls_b[3];
    for (int i = 0; i < 3; ++i) { cls_W[i] = (const float*)d_in[64 + i]; cls_b[i] = (const float*)d_in[67 + i]; }

    // -------- workspace layout --------
    float* ws = (float*)d_ws;
    size_t off = 0;
    auto alloc = [&](size_t n) { float* p = ws + off; off += (n + 255) & ~(size_t)255; return p; };
    const size_t NCtot = (size_t)NNODES * NC128;
    float* dis    = alloc(NNODES);
    float* colorH = alloc(NNODES);
    float* color[4];  for (int i = 0; i < 4; ++i) color[i] = alloc(NNODES);
    float* featW[4][4];                     // writable slots (featW[0][0] unused: == x)
    for (int j = 0; j < 4; ++j)
        for (int t = 0; t < 4; ++t)
            featW[j][t] = (j == 0 && t == 0) ? nullptr : alloc(NCtot);
    float* tmpH  = alloc(NCtot);
    float* tmpH2 = alloc(NCtot);
    int*   topIdx = (int*)alloc(4 * NB * TOPK);
    float* hbuf  = alloc((size_t)NB * 512);
    float* zbuf  = alloc((size_t)NB * 1536);
    float* z1    = alloc((size_t)NB * 128);
    float* z2    = alloc((size_t)NB * 128);

    const float* featC[4][4];
    for (int j = 0; j < 4; ++j) for (int t = 0; t < 4; ++t) featC[j][t] = featW[j][t];
    featC[0][0] = x;

    const int TB = 256;
    dim3 blk(TB);
    int gN   = (NNODES + TB - 1) / TB;
    int gE   = (NEDGE + TB - 1) / TB;
    int gNC  = (int)((NCtot + TB - 1) / TB);
    int gEs  = (int)(((size_t)NEDGE * 32 + TB - 1) / TB);
    int gGemm = ((NNODES / 16) + 3) / 4;   // 3125 tiles, 4 waves/block

    // degrees -> dis
    k_zero<<<gN, blk, 0, stream>>>(dis, NNODES);
    k_deg<<<gE, blk, 0, stream>>>(edge_dst, dis, NEDGE);
    k_rsqrt<<<gN, blk, 0, stream>>>(dis, NNODES);

    // 4 blocks x 3 GCN layers (+ trans projection between blocks)
    const float* bx = x;
    for (int now = 0; now < 4; ++now) {
        const float* g = bx;
        for (int i = 0; i < 3; ++i) {
            float* outp = featW[i + 1][now];
            k_gemm128<<<gGemm, 128, 0, stream>>>(g, gcn_W[now][i], tmpH, NNODES);
            k_gcn_init<<<gNC, blk, 0, stream>>>(tmpH, dis, gcn_b[now][i], outp, (int)NCtot);
            k_scatter<<<gEs, blk, 0, stream>>>(tmpH, edge_src, edge_dst, dis, outp, NEDGE);
            k_tanh<<<gNC, blk, 0, stream>>>(outp, (int)NCtot);
            g = outp;
        }
        if (now < 3) {
            k_gemm128<<<gGemm, 128, 0, stream>>>(bx, Wv[now], tmpH2, NNODES);
            k_gemm128<<<gGemm, 128, 0, stream>>>(tmpH2, Wo[now], featW[0][now + 1], NNODES);
            bx = featW[0][now + 1];
        }
    }

    // color GCNs (on each stream's last feature map)
    for (int j = 0; j < 4; ++j) {
        k_colorH<<<gN, blk, 0, stream>>>(featC[j][3], color_W[j], colorH, NNODES);
        k_color_init<<<gN, blk, 0, stream>>>(colorH, dis, color_b[j], color[j], NNODES);
        k_color_scatter<<<gE, blk, 0, stream>>>(colorH, edge_src, edge_dst, dis, color[j], NEDGE);
        k_tanh<<<gN, blk, 0, stream>>>(color[j], NNODES);
    }

    // sort-pool + conv tail per part
    FeatPtrs P;
    for (int j = 0; j < 4; ++j) {
        for (int t = 0; t < 4; ++t) P.f[j * 4 + t] = featC[j][t];
        P.c[j] = color[j];
    }
    for (int i = 0; i < 4; ++i) {
        k_topk<<<NB, 512, 0, stream>>>(color[i], topIdx + i * NB * TOPK);
        k_einsum<<<NB, 512, 0, stream>>>(P, topIdx + i * NB * TOPK, conv_W1[i], conv_b1[i],
                                         hbuf, i + 1, 513 * (i + 1));
        k_conv<<<NB, 512, 0, stream>>>(hbuf, conv_W2[i], conv_b2[i], zbuf, i);
    }

    // classifier + loss
    float* logits = (float*)d_out;
    k_dense<<<(NB * 128 + TB - 1) / TB, blk, 0, stream>>>(zbuf, cls_W[0], cls_b[0], z1, NB, 1536, 128, 2);
    k_dense<<<(NB * 128 + TB - 1) / TB, blk, 0, stream>>>(z1, cls_W[1], cls_b[1], z2, NB, 128, 128, 2);
    k_dense<<<(NB * 10 + TB - 1) / TB, blk, 0, stream>>>(z2, cls_W[2], cls_b[2], logits, NB, 128, 10, 1);
    k_loss<<<1, 128, 0, stream>>>(logits, y, logits + NB * 10);
}